// StarGateModel_86955907875132
// MI455X (gfx1250) — compile-verified
//
#include <hip/hip_runtime.h>
#include <math.h>

// ---------------------------------------------------------------------------
// StarGate model, CDNA5 (gfx1250) implementation.
// All GEMMs run on V_WMMA_F32_16X16X4_F32 (exact fp32 matrix path, wave32).
// Rows are partitioned by domain so domain-specific GEMMs run only on the
// rows that need them. Main GEMM: 256x64 block tile, 32-row wave strips
// (B fragments reused across 2 M-tiles), double-buffered LDS with
// register-staged global prefetch, LDS layouts chosen so every WMMA fragment
// is one aligned ds_load_b64 into an even VGPR pair (pad=20 avoids bank
// conflicts between half-waves).
// ---------------------------------------------------------------------------

typedef __attribute__((ext_vector_type(2))) float v2f;
typedef __attribute__((ext_vector_type(4))) float v4f;
typedef __attribute__((ext_vector_type(8))) float v8f;

#define NTHREADS 256
#define TM  256         // main GEMM block rows (8 waves x 32)
#define BN  64          // block cols (4 WMMA n-tiles)
#define BK  16          // k-chunk (4 WMMA k-steps)
#define BKP 20          // padded k-pitch: 8B-aligned pairs, conflict-free
#define FM  128         // film/final block rows (8 waves x 16)
#define NB  16384       // batch
#define ND  8           // domains

__device__ __forceinline__ float sigmoidf_(float v) {
    return 1.0f / (1.0f + __expf(-v));
}

__device__ __forceinline__ v8f wmma4(v2f a, v2f b, v8f c) {
    return __builtin_amdgcn_wmma_f32_16x16x4_f32(
        false, a, false, b, (short)0, c, false, false);
}

// Epilogues: 0 = linear(+bias), 1 = relu(+bias), 2 = gate: x * sigmoid(v+bias)
// SEG: per-domain segment (grid.z = domain, compact C rows, per-domain B/bias)
// GATHER_A: A rows fetched through rowsmap (compact -> original index)
template<int EP, bool SEG, bool GATHER_A>
__global__ __launch_bounds__(NTHREADS)
void wmma_gemm(const float* __restrict__ A,
               const float* __restrict__ Bw,
               const float* __restrict__ bias,
               const float* __restrict__ aux,   // gate epilogue: x
               float* __restrict__ C,
               int N, int K,
               const int* __restrict__ rowsmap,
               const int* __restrict__ counts,
               const int* __restrict__ segOff)
{
    __shared__ float As[2][TM][BKP];     // [buf][m][k]
    __shared__ float Bs[2][BN][BKP];     // [buf][n][k]  (transposed)
    __shared__ int   rowIdx[TM];

    const int tid     = threadIdx.x;
    const int lane    = tid & 31;
    const int wv      = tid >> 5;            // wave 0..7 -> 32-row strip
    const int colBase = blockIdx.y * BN;
    const int z       = SEG ? blockIdx.z : 0;

    int base = blockIdx.x * TM;
    int mrem = TM;
    int so   = 0;
    if (SEG) {
        const int cnt = counts[z];
        if (base >= cnt) return;             // block-uniform: EXEC stays full
        so   = segOff[z];
        mrem = cnt - base; if (mrem > TM) mrem = TM;
        Bw   += (size_t)z * (size_t)K * (size_t)N;
        bias += (size_t)z * (size_t)N;
    }

    {
        const int rc = (tid < mrem) ? tid : (mrem - 1);
        int g;
        if (GATHER_A)  g = rowsmap[so + base + rc];
        else if (SEG)  g = so + base + rc;
        else           g = base + tid;
        rowIdx[tid] = g;
    }
    __syncthreads();

    const int ar  = tid >> 2,  ac = (tid & 3) << 2;   // A chunk: 4 x float4
    const int rb  = tid >> 4,  cb = (tid & 15) << 2;  // B chunk: 1 x float4
    const int n16 = lane & 15;
    const int kp  = (lane >> 4) << 1;                 // k-pair per half-wave

    size_t aBase[4];
#pragma unroll
    for (int h = 0; h < 4; ++h)
        aBase[h] = (size_t)rowIdx[ar + h * 64] * (size_t)K + (size_t)ac;
    const size_t bBase = (size_t)rb * N + (size_t)(colBase + cb);

    v8f acc[2][4];
#pragma unroll
    for (int m = 0; m < 2; ++m)
#pragma unroll
        for (int t = 0; t < 4; ++t) acc[m][t] = 0.0f;

    // ---- prologue: stage chunk 0 ----
    v4f aReg[4], bReg;
#pragma unroll
    for (int h = 0; h < 4; ++h) aReg[h] = *(const v4f*)(A + aBase[h]);
    bReg = *(const v4f*)(Bw + bBase);
#pragma unroll
    for (int h = 0; h < 4; ++h) {
        float* d = &As[0][ar + h * 64][ac];
        d[0] = aReg[h][0]; d[1] = aReg[h][1];
        d[2] = aReg[h][2]; d[3] = aReg[h][3];
    }
#pragma unroll
    for (int j = 0; j < 4; ++j) Bs[0][cb + j][rb] = bReg[j];
    __syncthreads();

    int p = 0;
    for (int k0 = 0; k0 < K; k0 += BK) {
        const bool hasNext = (k0 + BK) < K;
        if (hasNext) {                        // prefetch next chunk to regs
#pragma unroll
            for (int h = 0; h < 4; ++h)
                aReg[h] = *(const v4f*)(A + aBase[h] + (size_t)(k0 + BK));
            bReg = *(const v4f*)(Bw + bBase + (size_t)(k0 + BK) * N);
        }
#pragma unroll
        for (int kk = 0; kk < BK; kk += 4) {
            const v2f a0 = *(const v2f*)&As[p][wv * 32 +      n16][kk + kp];
            const v2f a1 = *(const v2f*)&As[p][wv * 32 + 16 + n16][kk + kp];
#pragma unroll
            for (int t = 0; t < 4; ++t) {
                const v2f b = *(const v2f*)&Bs[p][t * 16 + n16][kk + kp];
                acc[0][t] = wmma4(a0, b, acc[0][t]);
                acc[1][t] = wmma4(a1, b, acc[1][t]);
            }
        }
        if (hasNext) {                        // fill the other buffer
            const int q = p ^ 1;
#pragma unroll
            for (int h = 0; h < 4; ++h) {
                float* d = &As[q][ar + h * 64][ac];
                d[0] = aReg[h][0]; d[1] = aReg[h][1];
                d[2] = aReg[h][2]; d[3] = aReg[h][3];
            }
#pragma unroll
            for (int j = 0; j < 4; ++j) Bs[q][cb + j][rb] = bReg[j];
        }
        __syncthreads();
        p ^= 1;
    }

    const int hi = lane >> 4;
#pragma unroll
    for (int m = 0; m < 2; ++m) {
#pragma unroll
        for (int t = 0; t < 4; ++t) {
            const int col = colBase + t * 16 + n16;
            const float bv = bias[col];
#pragma unroll
            for (int r = 0; r < 8; ++r) {
                const int mb = wv * 32 + m * 16 + r + 8 * hi;
                if (SEG && mb >= mrem) continue;
                float v = acc[m][t][r] + bv;
                if (EP == 1) v = fmaxf(v, 0.0f);
                if (EP == 2) {
                    const float xv = aux[(size_t)(base + mb) * N + col];
                    v = xv * sigmoidf_(v);
                }
                const size_t crow = SEG ? (size_t)(so + base + mb)
                                        : (size_t)(base + mb);
                C[crow * (size_t)N + col] = v;
            }
        }
    }
}

// FiLM: gamma = hd2c @ gW[d] + gb[d], beta = hd2c @ bW[d] + bb[d],
// fused[orig] = gamma * h_center[orig] + beta  (scatter to original rows).
__global__ __launch_bounds__(NTHREADS)
void film_fuse_kernel(const float* __restrict__ hd2c,
                      const float* __restrict__ gW, const float* __restrict__ gb,
                      const float* __restrict__ bW, const float* __restrict__ bb,
                      const float* __restrict__ hcenter,
                      const int* __restrict__ rowsmap,
                      const int* __restrict__ counts,
                      const int* __restrict__ segOff,
                      float* __restrict__ hfused)
{
    const int N = 256, K = 128;
    __shared__ float As [FM][BKP];
    __shared__ float BsG[BN][BKP];       // transposed
    __shared__ float BsB[BN][BKP];       // transposed
    __shared__ int   origIdx[FM];

    const int tid     = threadIdx.x;
    const int lane    = tid & 31;
    const int wv      = tid >> 5;
    const int colBase = blockIdx.y * BN;
    const int z       = blockIdx.z;

    const int cnt = counts[z];
    int base = blockIdx.x * FM;
    if (base >= cnt) return;
    const int so = segOff[z];
    int mrem = cnt - base; if (mrem > FM) mrem = FM;

    if (tid < FM) {
        const int rc = (tid < mrem) ? tid : (mrem - 1);
        origIdx[tid] = rowsmap[so + base + rc];
    }
    __syncthreads();

    const float* gWd = gW + (size_t)z * K * N;
    const float* bWd = bW + (size_t)z * K * N;

    v8f accG[4], accB[4];
#pragma unroll
    for (int t = 0; t < 4; ++t) { accG[t] = 0.0f; accB[t] = 0.0f; }

    const int ar  = tid >> 2,  ac = (tid & 3) << 2;
    const int rb  = tid >> 4,  cb = (tid & 15) << 2;
    const int n16 = lane & 15;
    const int kp  = (lane >> 4) << 1;

    for (int k0 = 0; k0 < K; k0 += BK) {
#pragma unroll
        for (int h = 0; h < 2; ++h) {
            const int r  = ar + h * 64;
            const int rc = (r < mrem) ? r : (mrem - 1);
            const v4f v = *(const v4f*)(hd2c + (size_t)(so + base + rc) * K + (k0 + ac));
            float* d = &As[r][ac];
            d[0] = v[0]; d[1] = v[1]; d[2] = v[2]; d[3] = v[3];
        }
        {
            const v4f v = *(const v4f*)(gWd + (size_t)(k0 + rb) * N + (colBase + cb));
            const v4f u = *(const v4f*)(bWd + (size_t)(k0 + rb) * N + (colBase + cb));
#pragma unroll
            for (int j = 0; j < 4; ++j) { BsG[cb + j][rb] = v[j]; BsB[cb + j][rb] = u[j]; }
        }
        __syncthreads();
#pragma unroll
        for (int kk = 0; kk < BK; kk += 4) {
            const v2f a = *(const v2f*)&As[wv * 16 + n16][kk + kp];
#pragma unroll
            for (int t = 0; t < 4; ++t) {
                const v2f g = *(const v2f*)&BsG[t * 16 + n16][kk + kp];
                const v2f b = *(const v2f*)&BsB[t * 16 + n16][kk + kp];
                accG[t] = wmma4(a, g, accG[t]);
                accB[t] = wmma4(a, b, accB[t]);
            }
        }
        __syncthreads();
    }

    const int hi = lane >> 4;
#pragma unroll
    for (int t = 0; t < 4; ++t) {
        const int col = colBase + t * 16 + n16;
        const float gbv = gb[z * N + col];
        const float bbv = bb[z * N + col];
#pragma unroll
        for (int r = 0; r < 8; ++r) {
            const int mb = wv * 16 + r + 8 * hi;
            if (mb >= mrem) continue;
            const int orig = origIdx[mb];
            const float gam = accG[t][r] + gbv;
            const float bet = accB[t][r] + bbv;
            const size_t o = (size_t)orig * N + col;
            hfused[o] = gam * hcenter[o] + bet;
        }
    }
}

// Final MLP: f = relu(hfused @ fW1 + fb1); out = sigmoid(f @ fW2 + fb2)
__global__ __launch_bounds__(NTHREADS)
void final_kernel(const float* __restrict__ hfused,
                  const float* __restrict__ fW1, const float* __restrict__ fb1,
                  const float* __restrict__ fW2, const float* __restrict__ fb2,
                  float* __restrict__ out)
{
    const int N = 64, K = 256;
    __shared__ float As[FM][BKP];
    __shared__ float Bs[BN][BKP];        // transposed
    __shared__ float fbuf[FM][65];

    const int tid  = threadIdx.x;
    const int lane = tid & 31;
    const int wv   = tid >> 5;
    const int base = blockIdx.x * FM;

    v8f acc[4];
#pragma unroll
    for (int t = 0; t < 4; ++t) acc[t] = 0.0f;

    const int ar  = tid >> 2,  ac = (tid & 3) << 2;
    const int rb  = tid >> 4,  cb = (tid & 15) << 2;
    const int n16 = lane & 15;
    const int kp  = (lane >> 4) << 1;

    for (int k0 = 0; k0 < K; k0 += BK) {
#pragma unroll
        for (int h = 0; h < 2; ++h) {
            const int r = ar + h * 64;
            const v4f v = *(const v4f*)(hfused + (size_t)(base + r) * K + (k0 + ac));
            float* d = &As[r][ac];
            d[0] = v[0]; d[1] = v[1]; d[2] = v[2]; d[3] = v[3];
        }
        {
            const v4f v = *(const v4f*)(fW1 + (size_t)(k0 + rb) * N + cb);
#pragma unroll
            for (int j = 0; j < 4; ++j) Bs[cb + j][rb] = v[j];
        }
        __syncthreads();
#pragma unroll
        for (int kk = 0; kk < BK; kk += 4) {
            const v2f a = *(const v2f*)&As[wv * 16 + n16][kk + kp];
#pragma unroll
            for (int t = 0; t < 4; ++t) {
                const v2f b = *(const v2f*)&Bs[t * 16 + n16][kk + kp];
                acc[t] = wmma4(a, b, acc[t]);
            }
        }
        __syncthreads();
    }

    const int hi = lane >> 4;
#pragma unroll
    for (int t = 0; t < 4; ++t) {
        const int col = t * 16 + n16;
        const float bv = fb1[col];
#pragma unroll
        for (int r = 0; r < 8; ++r) {
            const int mb = wv * 16 + r + 8 * hi;
            fbuf[mb][col] = fmaxf(acc[t][r] + bv, 0.0f);
        }
    }
    __syncthreads();

    if (tid < FM) {
        float s = fb2[0];
#pragma unroll 8
        for (int i = 0; i < 64; ++i) s += fbuf[tid][i] * fW2[i];
        out[base + tid] = sigmoidf_(s);
    }
}

// ---------------- domain partitioning (counts -> prefix -> scatter) --------
__global__ void count_kernel(const long long* __restrict__ dom,
                             int* __restrict__ counts, int n)
{
    const int i = blockIdx.x * blockDim.x + threadIdx.x;
    if (i < n) atomicAdd(&counts[(int)dom[i]], 1);
}

__global__ void offsets_kernel(const int* __restrict__ counts,
                               int* __restrict__ segOff)
{
    if (threadIdx.x == 0 && blockIdx.x == 0) {
        int acc = 0;
        for (int d = 0; d < ND; ++d) { segOff[d] = acc; acc += counts[d]; }
    }
}

__global__ void scatter_kernel(const long long* __restrict__ dom,
                               const int* __restrict__ segOff,
                               int* __restrict__ fill,
                               int* __restrict__ rowsmap, int n)
{
    const int i = blockIdx.x * blockDim.x + threadIdx.x;
    if (i < n) {
        const int d = (int)dom[i];
        const int pos = segOff[d] + atomicAdd(&fill[d], 1);
        rowsmap[pos] = i;
    }
}

// ---------------------------------------------------------------------------
extern "C" void kernel_launch(void* const* d_in, const int* in_sizes, int n_in,
                              void* d_out, int out_size, void* d_ws, size_t ws_size,
                              hipStream_t stream)
{
    const float*     x      = (const float*)d_in[0];
    const long long* dom    = (const long long*)d_in[1];
    const float*     gate_W = (const float*)d_in[2];
    const float*     gate_b = (const float*)d_in[3];
    const float*     cW1    = (const float*)d_in[4];
    const float*     cb1    = (const float*)d_in[5];
    const float*     cW2    = (const float*)d_in[6];
    const float*     cb2    = (const float*)d_in[7];
    const float*     cW3    = (const float*)d_in[8];
    const float*     cb3    = (const float*)d_in[9];
    const float*     dW1    = (const float*)d_in[10];
    const float*     db1    = (const float*)d_in[11];
    const float*     dW2    = (const float*)d_in[12];
    const float*     db2    = (const float*)d_in[13];
    const float*     gW     = (const float*)d_in[14];
    const float*     gb     = (const float*)d_in[15];
    const float*     bW     = (const float*)d_in[16];
    const float*     bb     = (const float*)d_in[17];
    const float*     fW1    = (const float*)d_in[18];
    const float*     fb1    = (const float*)d_in[19];
    const float*     fW2    = (const float*)d_in[20];
    const float*     fb2    = (const float*)d_in[21];
    float*           out    = (float*)d_out;

    char* ws = (char*)d_ws;
    const size_t MB = 1ull << 20;
    float* xg     = (float*)(ws + 0);            // 64 MB  [B,1024]
    float* h1     = (float*)(ws + 64 * MB);      // 64 MB  [B,1024] (recycled)
    float* h2     = (float*)(ws + 128 * MB);     // 32 MB  [B,512]
    float* hc     = (float*)(ws + 160 * MB);     // 16 MB  [B,256]
    float* hd1c   = (float*)(ws + 64 * MB);      // 16 MB  compact [B,256]
    float* hd2c   = (float*)(ws + 80 * MB);      //  8 MB  compact [B,128]
    float* hfused = (float*)(ws + 88 * MB);      // 16 MB  [B,256]
    int*   meta   = (int*)(ws + 176 * MB);
    int*   counts = meta;          // 8
    int*   segOff = meta + 8;      // 8
    int*   fill   = meta + 16;     // 8
    int*   rowsmap= meta + 32;     // 16384

    const dim3 blk(NTHREADS);

    // Partition rows by domain (output is order-independent: each row's math
    // is independent and results scatter back by original index).
    hipMemsetAsync(meta, 0, 32 * sizeof(int), stream);
    count_kernel  <<<NB / NTHREADS, blk, 0, stream>>>(dom, counts, NB);
    offsets_kernel<<<1, 1, 0, stream>>>(counts, segOff);
    scatter_kernel<<<NB / NTHREADS, blk, 0, stream>>>(dom, segOff, fill, rowsmap, NB);

    // GateNU: xg = x * sigmoid(x @ gate_W + gate_b)
    wmma_gemm<2, false, false><<<dim3(NB / TM, 1024 / BN, 1), blk, 0, stream>>>(
        x, gate_W, gate_b, x, xg, 1024, 1024, nullptr, nullptr, nullptr);
    // Center net
    wmma_gemm<1, false, false><<<dim3(NB / TM, 1024 / BN, 1), blk, 0, stream>>>(
        xg, cW1, cb1, nullptr, h1, 1024, 1024, nullptr, nullptr, nullptr);
    wmma_gemm<1, false, false><<<dim3(NB / TM, 512 / BN, 1), blk, 0, stream>>>(
        h1, cW2, cb2, nullptr, h2, 512, 1024, nullptr, nullptr, nullptr);
    wmma_gemm<0, false, false><<<dim3(NB / TM, 256 / BN, 1), blk, 0, stream>>>(
        h2, cW3, cb3, nullptr, hc, 256, 512, nullptr, nullptr, nullptr);
    // Domain nets: only the rows of each domain (gathered A, compact output)
    wmma_gemm<1, true, true><<<dim3(NB / TM, 256 / BN, ND), blk, 0, stream>>>(
        xg, dW1, db1, nullptr, hd1c, 256, 1024, rowsmap, counts, segOff);
    wmma_gemm<0, true, false><<<dim3(NB / TM, 128 / BN, ND), blk, 0, stream>>>(
        hd1c, dW2, db2, nullptr, hd2c, 128, 256, rowsmap, counts, segOff);
    // FiLM gamma/beta + fuse with h_center, scattered back to original rows
    film_fuse_kernel<<<dim3(NB / FM, 256 / BN, ND), blk, 0, stream>>>(
        hd2c, gW, gb, bW, bb, hc, rowsmap, counts, segOff, hfused);
    // Final MLP + sigmoid
    final_kernel<<<dim3(NB / FM, 1, 1), blk, 0, stream>>>(
        hfused, fW1, fb1, fW2, fb2, out);
}